// Group_51058571215371
// MI455X (gfx1250) — compile-verified
//
#include <hip/hip_runtime.h>

typedef float v2f __attribute__((ext_vector_type(2)));
typedef float v8f __attribute__((ext_vector_type(8)));

#define B_  8
#define N_  8192
#define M_  2048
#define C_  128
#define K_  16
#define CHUNK 2048      // points staged per LDS pass (32 KB as float4)
#define WGQ 128         // queries per workgroup (8 waves * 16)
#define NT1 256

// ---------------------------------------------------------------------------
// Kernel 1: fused distance-score WMMA + per-query top-16 selection.
// score(n,m) = |p_n|^2 - 2 p_n . q_m   (rank-equivalent to squared L2 dist,
// since |q_m|^2 is constant per query column).
// One wave owns 16 query columns; V_WMMA_F32_16X16X4_F32 produces a full
// 16x16 score tile per iteration from an LDS-staged A fragment.
// Top-16 kept per lane as an ascending register list; inserts use a linear
// min/max carry cascade (1 cmp + 4 cndmask per slot), guarded first by a
// per-tile min-of-8 test so non-contributing tiles cost ~10 VALU ops.
// ---------------------------------------------------------------------------
__global__ __launch_bounds__(NT1) void knn_topk_wmma(
    const float* __restrict__ points,      // (B,3,N)
    const float* __restrict__ new_points,  // (B,3,M)
    int* __restrict__ idx_out)             // (B,M,K)
{
  __shared__ float4 sp[CHUNK];             // {x,y,z,|p|^2} interleaved

  const int b    = blockIdx.x >> 4;        // 16 m-blocks per batch
  const int mblk = blockIdx.x & 15;
  const int t    = threadIdx.x;
  const int wave = t >> 5;
  const int lane = t & 31;
  const int col  = lane & 15;
  const int m    = mblk * WGQ + wave * 16 + col;

  const float* pb = points     + (size_t)b * 3 * N_;
  const float* qb = new_points + (size_t)b * 3 * M_;

  // B fragment (4x16): lanes 0-15 carry K0,K1; lanes 16-31 carry K2,K3.
  const float qx = qb[m], qy = qb[M_ + m], qz = qb[2 * M_ + m];
  v2f bfrag;
  bfrag[0] = (lane < 16) ? (-2.0f * qx) : (-2.0f * qz);
  bfrag[1] = (lane < 16) ? (-2.0f * qy) : 1.0f;

  // Per-lane top-16 (ascending by score); lane L covers tile rows 0-7,
  // lane L+16 covers rows 8-15 of the same query column.
  float tv[K_];
  int   ti[K_];
#pragma unroll
  for (int j = 0; j < K_; ++j) { tv[j] = __builtin_inff(); ti[j] = 0x7fffffff; }

  const int rowoff = (lane >> 4) * 8;
  const float* spf = (const float*)sp;
  const float* ap0 = spf + col * 4 + ((lane < 16) ? 0 : 2);

  for (int c0 = 0; c0 < N_; c0 += CHUNK) {
    // Stage chunk: coalesced global reads, compute |p|^2 once.
    for (int e = t; e < CHUNK; e += NT1) {
      const int n = c0 + e;
      const float x = pb[n], y = pb[N_ + n], z = pb[2 * N_ + n];
      sp[e] = make_float4(x, y, z, x * x + y * y + z * z);
    }
    __syncthreads();

    for (int nn = 0; nn < CHUNK; nn += 16) {
      // A fragment (16x4): one 8B LDS load per lane, conflict-free.
      v2f afrag = *(const v2f*)(ap0 + nn * 4);
      v8f acc = {};
      acc = __builtin_amdgcn_wmma_f32_16x16x4_f32(
          false, afrag, false, bfrag, (short)0, acc, false, false);

      // Per-tile guard: min of the 8 candidates (branch-free tree).
      float t01 = fminf(acc[0], acc[1]);
      float t23 = fminf(acc[2], acc[3]);
      float t45 = fminf(acc[4], acc[5]);
      float t67 = fminf(acc[6], acc[7]);
      float tmin = fminf(fminf(t01, t23), fminf(t45, t67));

      if (tmin < tv[K_ - 1]) {
        const int nbase = c0 + nn + rowoff;
#pragma unroll
        for (int j = 0; j < 8; ++j) {
          float cv = acc[j];
          if (cv < tv[K_ - 1]) {            // strict <: older (lower n) wins ties
            int ci = nbase + j;
            // Linear carry cascade: insert (cv,ci), drop previous worst.
#pragma unroll
            for (int r = 0; r < K_; ++r) {
              const bool sw = cv < tv[r];
              const float vlo = sw ? cv    : tv[r];
              const float vhi = sw ? tv[r] : cv;
              const int   ilo = sw ? ci    : ti[r];
              const int   ihi = sw ? ti[r] : ci;
              tv[r] = vlo; ti[r] = ilo;
              cv = vhi;    ci = ihi;
            }
          }
        }
      }
    }
    __syncthreads();
  }

  // Merge the two half-lists of each query column (lanes L and L+16):
  // take the 16 smallest of 32, then sort the bitonic result.
  float pv[K_]; int pi[K_];
#pragma unroll
  for (int j = 0; j < K_; ++j) {
    pv[j] = __shfl_xor(tv[j], 16, 32);
    pi[j] = __shfl_xor(ti[j], 16, 32);
  }
  float mv[K_]; int mi[K_];
#pragma unroll
  for (int j = 0; j < K_; ++j) {
    const float av = tv[j], bv = pv[K_ - 1 - j];
    const int   ai = ti[j], bi = pi[K_ - 1 - j];
    const bool less = (av < bv) || (av == bv && ai < bi);
    mv[j] = less ? av : bv;
    mi[j] = less ? ai : bi;
  }
#pragma unroll
  for (int stride = 8; stride >= 1; stride >>= 1) {
#pragma unroll
    for (int j = 0; j < K_; ++j) {
      if ((j & stride) == 0) {
        const int j2 = j + stride;
        const bool sw = (mv[j2] < mv[j]) || (mv[j2] == mv[j] && mi[j2] < mi[j]);
        const float fv = sw ? mv[j2] : mv[j];
        const float fw = sw ? mv[j]  : mv[j2];
        const int   iv = sw ? mi[j2] : mi[j];
        const int   iw = sw ? mi[j]  : mi[j2];
        mv[j] = fv; mv[j2] = fw;
        mi[j] = iv; mi[j2] = iw;
      }
    }
  }

  if (lane < 16) {
    int* op = idx_out + ((size_t)b * M_ + m) * K_;
#pragma unroll
    for (int j = 0; j < K_; ++j) op[j] = mi[j];
  }
}

// ---------------------------------------------------------------------------
// Kernel 2: gather + concat. One 128-thread block per (b,m).
// Writes are k-contiguous (64B per channel per 16 lanes); scattered feature
// reads stay L2-resident (feature set is 33 MB << 192 MB L2).
// ---------------------------------------------------------------------------
__global__ __launch_bounds__(128) void group_gather(
    const float* __restrict__ points,      // (B,3,N)
    const float* __restrict__ new_points,  // (B,3,M)
    const float* __restrict__ features,    // (B,C,N)
    const int*   __restrict__ idxws,       // (B,M,K)
    float*       __restrict__ out)         // (B,3+C,M,K)
{
  const int bm = blockIdx.x;
  const int b  = bm >> 11;                 // / 2048
  const int m  = bm & (M_ - 1);
  const int t  = threadIdx.x;

  __shared__ int sidx[K_];
  if (t < K_) sidx[t] = idxws[(size_t)bm * K_ + t];
  __syncthreads();

  const int k = t & 15;
  const int n = sidx[k];
  const size_t ob = (size_t)b * (3 + C_) * M_ * K_;

  if (t < 48) {
    const int c = t >> 4;                  // 0..2
    const float pvv = points[(size_t)b * 3 * N_ + (size_t)c * N_ + n];
    const float qvv = new_points[(size_t)b * 3 * M_ + (size_t)c * M_ + m];
    out[ob + (size_t)c * M_ * K_ + (size_t)m * K_ + k] = pvv - qvv;
  }

  const float* fb = features + (size_t)b * C_ * N_;
  for (int c = (t >> 4); c < C_; c += 8) {
    out[ob + (size_t)(3 + c) * M_ * K_ + (size_t)m * K_ + k] =
        fb[(size_t)c * N_ + n];
  }
}

extern "C" void kernel_launch(void* const* d_in, const int* in_sizes, int n_in,
                              void* d_out, int out_size, void* d_ws, size_t ws_size,
                              hipStream_t stream) {
  (void)in_sizes; (void)n_in; (void)out_size; (void)ws_size;
  const float* points     = (const float*)d_in[0];
  const float* new_points = (const float*)d_in[1];
  const float* features   = (const float*)d_in[2];
  // d_in[3] = num_samples (fixed at 16)

  int* idxws = (int*)d_ws;                 // B*M*K ints = 1 MB

  knn_topk_wmma<<<dim3(B_ * (M_ / WGQ)), dim3(NT1), 0, stream>>>(
      points, new_points, idxws);
  group_gather<<<dim3(B_ * M_), dim3(128), 0, stream>>>(
      points, new_points, features, idxws, (float*)d_out);
}